// LIFNode_5102421147975
// MI455X (gfx1250) — compile-verified
//
#include <hip/hip_runtime.h>

// ---------------------------------------------------------------------------
// Multi-step LIF (TAU=2, V_TH=1, V_RESET=0.5), x:(T,B,N) f32 -> spikes f32.
// Memory-bound scan: 512 MB traffic -> ~22us floor @ 23.3 TB/s.
// Strategy: 8-deep async global->LDS pipeline (ASYNCcnt), ds_load_b128
// consumption, non-temporal b128 spike stores.
// ---------------------------------------------------------------------------

typedef float v4f __attribute__((ext_vector_type(4)));
typedef int   v4i __attribute__((ext_vector_type(4)));

typedef v4i __attribute__((address_space(1)))* as1_v4i_p;
typedef v4i __attribute__((address_space(3)))* as3_v4i_p;

#ifndef __has_builtin
#define __has_builtin(x) 0
#endif

constexpr int kT      = 64;               // timesteps
constexpr int kBlock  = 256;              // 8 waves
constexpr int kVec    = 4;                // floats per thread per step
constexpr int kTile   = kBlock * kVec;    // 1024 floats = 4KB per stage
constexpr int kStages = 8;                // 32KB LDS, 8 async loads in flight

__device__ __forceinline__ void async_load16(const float* g, float* l) {
#if __has_builtin(__builtin_amdgcn_global_load_async_to_lds_b128)
  __builtin_amdgcn_global_load_async_to_lds_b128(
      (as1_v4i_p)(unsigned long long)g,   // global src (AS1 int4*)
      (as3_v4i_p)l,                       // LDS dst (AS3 int4*)
      /*offset=*/0, /*cpol=*/0);
#else
  unsigned laddr = (unsigned)(unsigned long long)(as3_v4i_p)l;
  asm volatile("global_load_async_to_lds_b128 %0, %1, off"
               :: "v"(laddr), "v"(g)
               : "memory");
#endif
}

template <int N>
__device__ __forceinline__ void wait_async() {
  asm volatile("" ::: "memory");
#if __has_builtin(__builtin_amdgcn_s_wait_asynccnt)
  __builtin_amdgcn_s_wait_asynccnt(N);
#else
  asm volatile("s_wait_asynccnt %0" :: "i"(N));
#endif
  asm volatile("" ::: "memory");
}

__device__ __forceinline__ void wait_ds0() {
  // Make slot overwrite by the async engine safe vs. an in-flight ds_load.
  asm volatile("s_wait_dscnt 0x0" ::: "memory");
}

// One LIF step on 4 lanes-worth of data. Mirrors reference expression order:
//   h = (v - (v - 0.5)/2) + x ; s = h>=1 ; v' = s ? 0.5 : h
__device__ __forceinline__ v4f lif_step(v4f& v, v4f xv) {
  v4f o;
#pragma unroll
  for (int i = 0; i < 4; ++i) {
    float h  = (v[i] - (v[i] - 0.5f) * 0.5f) + xv[i];
    bool  sp = h >= 1.0f;
    o[i] = sp ? 1.0f : 0.0f;
    v[i] = sp ? 0.5f : h;
  }
  return o;
}

__global__ __launch_bounds__(kBlock) void lif_async_kernel(
    const float* __restrict__ x, float* __restrict__ out, int bn) {
  __shared__ float stages[kStages * kTile];   // 32 KB

  const int tid  = threadIdx.x;
  const long long base = (long long)blockIdx.x * kTile + (long long)tid * kVec;
  const float* gx   = x   + base;
  float*       gout = out + base;
  float*       lmine = &stages[tid * kVec];   // this thread's 16B per stage

  // Prologue: fill all pipeline stages (stages 0..7 == timesteps 0..7).
#pragma unroll
  for (int s = 0; s < kStages; ++s)
    async_load16(gx + (long long)s * bn, lmine + s * kTile);

  const float* gnext = gx + (long long)kStages * bn;

  v4f v = {0.5f, 0.5f, 0.5f, 0.5f};   // V_RESET

  // Main: timesteps 0 .. T-kStages-1  (56 = 7 groups of 8 so slot is constant)
  for (int grp = 0; grp < (kT - kStages) / kStages; ++grp) {
#pragma unroll
    for (int k = 0; k < kStages; ++k) {
      wait_async<kStages - 1>();               // oldest stage (t) is in LDS
      v4f xv = *(const v4f*)(lmine + k * kTile);  // ds_load_b128
      v4f o  = lif_step(v, xv);
      __builtin_nontemporal_store(o, (v4f*)gout); // write-once spikes: NT
      wait_ds0();                              // slot read fully retired
      async_load16(gnext, lmine + k * kTile);  // refill slot with t+8
      gnext += bn;
      gout  += bn;
    }
  }

  // Tail: last kStages timesteps, everything already issued.
  wait_async<0>();
#pragma unroll
  for (int k = 0; k < kStages; ++k) {
    v4f xv = *(const v4f*)(lmine + k * kTile);
    v4f o  = lif_step(v, xv);
    __builtin_nontemporal_store(o, (v4f*)gout);
    gout += bn;
  }
}

// Generic fallback for any column remainder (not hit for B*N = 1,048,576).
__global__ void lif_tail_kernel(const float* __restrict__ x,
                                float* __restrict__ out, int bn, int start) {
  int idx = start + blockIdx.x * blockDim.x + threadIdx.x;
  if (idx >= bn) return;
  float v = 0.5f;
  for (int t = 0; t < kT; ++t) {
    float h  = (v - (v - 0.5f) * 0.5f) + x[(long long)t * bn + idx];
    bool  sp = h >= 1.0f;
    out[(long long)t * bn + idx] = sp ? 1.0f : 0.0f;
    v = sp ? 0.5f : h;
  }
}

extern "C" void kernel_launch(void* const* d_in, const int* in_sizes, int n_in,
                              void* d_out, int out_size, void* d_ws, size_t ws_size,
                              hipStream_t stream) {
  (void)n_in; (void)out_size; (void)d_ws; (void)ws_size;
  const float* x   = (const float*)d_in[0];
  float*       out = (float*)d_out;

  const long long total = (long long)in_sizes[0];
  const int bn = (int)(total / kT);        // B*N columns

  const int nblk      = bn / kTile;        // full 4KB-tile blocks
  const int rem_start = nblk * kTile;
  const int rem       = bn - rem_start;

  if (nblk > 0)
    lif_async_kernel<<<nblk, kBlock, 0, stream>>>(x, out, bn);
  if (rem > 0)
    lif_tail_kernel<<<(rem + 255) / 256, 256, 0, stream>>>(x, out, bn, rem_start);
}